// sampling_16260746183117
// MI455X (gfx1250) — compile-verified
//
#include <hip/hip_runtime.h>
#include <stdint.h>

typedef __attribute__((ext_vector_type(2))) float v2f;
typedef __attribute__((ext_vector_type(8))) float v8f;

#define BATCH   32
#define N_IN    20480
#define N_OUT   5120
#define FDIM    128
#define RB      16          // output rows per block
#define KTOT    48          // RB * 3 nnz
#define NCHUNK  12          // KTOT / 4  (K-steps of the 16x16x4 f32 WMMA)

__global__ __launch_bounds__(256)
void coma_downsample_wmma(const float* __restrict__ x,
                          const int*   __restrict__ cols,
                          const float* __restrict__ vals,
                          float*       __restrict__ out)
{
    __shared__ float s_x[KTOT][FDIM];   // 24 KB: gathered input rows
    __shared__ float s_val[KTOT];
    __shared__ int   s_col[KTOT];

    const int t    = threadIdx.x;
    const int lane = t & 31;
    const int wv   = t >> 5;            // wave id 0..7  -> f-tile
    const int rblk = blockIdx.x;        // 0..319
    const int b    = blockIdx.y;        // 0..31
    const int r0   = rblk * RB;

    // ---- stage the 48 (col, val) pairs for this row block ----
    if (t < KTOT) {
        s_col[t] = cols[r0 * 3 + t];
        s_val[t] = vals[r0 * 3 + t];
    }
    __syncthreads();

    // ---- async gather: 48 rows x 512B from global -> LDS ----
    // Each wave issues 6 rows; every row is one fully-coalesced 512B burst
    // (32 lanes x b128), streamed straight into LDS (no VGPR round trip).
    const float* xb = x + (size_t)b * N_IN * FDIM;
    #pragma unroll
    for (int i = 0; i < 6; ++i) {
        const int row = wv * 6 + i;
        unsigned voff = (unsigned)s_col[row] * (FDIM * 4) + (unsigned)lane * 16u;
        unsigned lds  = (unsigned)(uintptr_t)(&s_x[row][lane * 4]);
        asm volatile("global_load_async_to_lds_b128 %0, %1, %2"
                     :: "v"(lds), "v"(voff), "s"(xb)
                     : "memory");
    }
    asm volatile("s_wait_asynccnt 0" ::: "memory");
    __syncthreads();

    // ---- WMMA: out_tile(16x16) = A(16x48) x B(48x16), 12 K-chunks of 4 ----
    // A[i][k] = vals[k] if k/3 == i else 0   (rows sorted, 3 nnz each)
    // B[k][n] = s_x[k][f0 + n]
    //
    // Lane (half = lane>>4, m = lane&15) holds, for chunk c:
    //   A slots K = k0, k0+1 with k0 = 4c + 2*half  (A: M=m)
    //   B slots K = k0, k0+1 at column f0+m
    // All per-chunk LDS offsets from the lane-constant bases are
    // compile-time constants -> no address math inside the loop.
    // A's sparsity test k/3==m is (unsigned)(k - 3m) < 3 with
    // dbase = 2*half - 3*m hoisted -> one add + cmp + cndmask per element.
    const int half = lane >> 4;
    const int m    = lane & 15;
    const int f0   = wv * 16;

    const float* bp    = &s_x[2 * half][f0 + m];   // B base (this lane)
    const float* vp    = &s_val[2 * half];         // vals base (this lane)
    const int    dbase = 2 * half - 3 * m;

    v8f acc = {};                   // C starts at zero
    #pragma unroll
    for (int c = 0; c < NCHUNK; ++c) {
        v2f a, bb;
        const float va0 = vp[c * 4];               // one ds_load_b64 pair
        const float va1 = vp[c * 4 + 1];
        const int   d0  = dbase + c * 4;
        a.x = ((unsigned)d0       < 3u) ? va0 : 0.0f;
        a.y = ((unsigned)(d0 + 1) < 3u) ? va1 : 0.0f;
        bb.x = bp[(c * 4    ) * FDIM];             // const offs: c*2048
        bb.y = bp[(c * 4 + 1) * FDIM];             // and c*2048+512
        acc = __builtin_amdgcn_wmma_f32_16x16x4_f32(
                  /*neg_a=*/false, a, /*neg_b=*/false, bb,
                  /*c_mod=*/(short)0, acc,
                  /*reuse_a=*/false, /*reuse_b=*/false);
    }

    // ---- store D: VGPR g holds out[M = g + 8*half][N = m] of the tile ----
    // Non-temporal: output is write-once, keep L2 for the gather working set.
    float* ob = out + ((size_t)b * N_OUT + r0) * FDIM + f0 + m;
    #pragma unroll
    for (int g = 0; g < 8; ++g) {
        const int row = g + half * 8;
        __builtin_nontemporal_store(acc[g], ob + (size_t)row * FDIM);
    }
}

extern "C" void kernel_launch(void* const* d_in, const int* in_sizes, int n_in,
                              void* d_out, int out_size, void* d_ws, size_t ws_size,
                              hipStream_t stream)
{
    const float* x    = (const float*)d_in[0];
    // d_in[1] = D_rows: unused (rows are sorted with exactly 3 nnz per row)
    const int*   cols = (const int*)d_in[2];
    const float* vals = (const float*)d_in[3];
    float*       out  = (float*)d_out;

    dim3 grid(N_OUT / RB, BATCH);   // (320, 32)
    coma_downsample_wmma<<<grid, 256, 0, stream>>>(x, cols, vals, out);
}